// Image_Direct_AttenwMask_Sampler_89481348644939
// MI455X (gfx1250) — compile-verified
//
#include <hip/hip_runtime.h>

// ---------------------------------------------------------------------------
// CDNA5 / gfx1250 implementation of Image_Direct_AttenwMask_Sampler.
// All GEMM-shaped math goes through V_WMMA_F32_16X16X4_F32 (fp32 matrix core).
// GEMM + attention loops are software-pipelined (depth-1 prefetch) so the
// backend can issue s_wait_loadcnt with loads still in flight instead of the
// load -> s_wait_loadcnt 0 -> wmma lockstep seen in the previous round.
// ---------------------------------------------------------------------------

typedef float v2f __attribute__((ext_vector_type(2)));
typedef float v8f __attribute__((ext_vector_type(8)));

#define RESO 32
#define VIT 16
#define NQ 3072        // queries per batch
#define NK 257         // key tokens per batch
#define BATCH 8
#define E 128
#define TRI_C 64
#define IMG_C 1280
#define OUT_C 64
#define HEADS 8
#define DH 16

__device__ __forceinline__ v8f v8zero() {
  v8f z = {0.f, 0.f, 0.f, 0.f, 0.f, 0.f, 0.f, 0.f};
  return z;
}

// D = A(16x4,f32) x B(4x16,f32) + C(16x16,f32)
__device__ __forceinline__ v8f wmma4(v2f a, v2f b, v8f c) {
  return __builtin_amdgcn_wmma_f32_16x16x4_f32(false, a, false, b, (short)0, c,
                                               false, false);
}

// ---------------------------------------------------------------------------
// Sinusoidal positional encoding: pe[p][d] = sin/cos(p * 10000^(-2i/E))
// ---------------------------------------------------------------------------
__global__ void pe_kernel(float* __restrict__ pe, int length) {
  int id = blockIdx.x * 256 + threadIdx.x;
  if (id >= length * E) return;
  int p = id >> 7;
  int d = id & 127;
  // ln(10000) = 9.210340371976184
  float freq = expf(-(float)(d & ~1) * (9.210340371976184f / 128.0f));
  float ang = (float)p * freq;
  pe[id] = (d & 1) ? cosf(ang) : sinf(ang);
}

// ---------------------------------------------------------------------------
// tf[b][n][c] = triplane[b][c][n]   (B,64,3072) -> (B,3072,64)
// ---------------------------------------------------------------------------
__global__ void transpose_tf(const float* __restrict__ tri,
                             float* __restrict__ tf) {
  int id = blockIdx.x * 256 + threadIdx.x;  // < B*64*3072
  int b = id / (TRI_C * NQ);
  int r = id - b * (TRI_C * NQ);
  int c = r / NQ;
  int n = r - c * NQ;
  tf[((size_t)b * NQ + n) * TRI_C + c] = tri[id];
}

// ---------------------------------------------------------------------------
// Projective visibility mask -> 256-bit bitmask per (b, query token).
// Bit m set => key token (m+1) is VISIBLE. Key 0 is always visible.
// A depth sample's radius-1 disk covers <= 9 integer ViT cells, so we
// rasterize instead of building the (B,98304,256) distance matrix.
// ---------------------------------------------------------------------------
__global__ __launch_bounds__(256) void mask_kernel(
    const float* __restrict__ proj, unsigned* __restrict__ maskb) {
  const int b = blockIdx.y;
  const int q = blockIdx.x * 256 + threadIdx.x;  // 0..3071
  float pm[12];
#pragma unroll
  for (int i = 0; i < 12; ++i) pm[i] = proj[b * 12 + i];

  const int plane = q >> 10;
  const int rem = q & 1023;
  const int ii = rem >> 5;
  const int jj = rem & 31;
  const float C1 = 2.0f * (1.0f + 0.1f + 1e-5f) / 31.0f;  // grid step
  const float C2 = 1.0f + 0.1f + 1e-5f;                   // half-extent
  const float xi = (float)ii * C1 - C2;
  const float xj = (float)jj * C1 - C2;

  unsigned bits[8] = {0u, 0u, 0u, 0u, 0u, 0u, 0u, 0u};
  for (int d = 0; d < RESO; ++d) {
    float xd = (float)d * C1 - C2;
    float cx, cy, cz;
    if (plane == 0) { cx = xj; cy = xd; cz = xi; }        // xzy
    else if (plane == 1) { cx = xj; cy = xi; cz = xd; }   // xyz
    else { cx = xd; cy = xj; cz = xi; }                   // yzx
    float c0 = pm[0] * cx + pm[1] * cy + pm[2] * cz + pm[3];
    float c1 = pm[4] * cx + pm[5] * cy + pm[6] * cz + pm[7];
    float c2 = pm[8] * cx + pm[9] * cy + pm[10] * cz + pm[11];
    float u = c0 / c2 / 223.0f;
    float v = c1 / c2 / 223.0f;
    if (u != u || v != v) continue;  // NaN -> contributes nothing (jnp semantics)
    u = fminf(fmaxf(u, 0.0f), 1.0f) * 15.0f;
    v = fminf(fmaxf(v, 0.0f), 1.0f) * 15.0f;
    int x0 = (int)ceilf(u - 1.0f); if (x0 < 0) x0 = 0;
    int x1 = (int)floorf(u + 1.0f); if (x1 > 15) x1 = 15;
    int y0 = (int)ceilf(v - 1.0f); if (y0 < 0) y0 = 0;
    int y1 = (int)floorf(v + 1.0f); if (y1 > 15) y1 = 15;
    for (int my = y0; my <= y1; ++my)
      for (int mx = x0; mx <= x1; ++mx) {
        float dx = u - (float)mx, dy = v - (float)my;
        if (dx * dx + dy * dy < 1.0f) {
          int m = my * 16 + mx;
          bits[m >> 5] |= (1u << (m & 31));
        }
      }
  }
  unsigned* out = maskb + ((size_t)b * NQ + q) * 8;
#pragma unroll
  for (int i = 0; i < 8; ++i) out[i] = bits[i];
}

// ---------------------------------------------------------------------------
// Fold the two output projections: W_final = Wp @ out_w (64x128),
// b_final = bp + Wp @ out_b (64).
// ---------------------------------------------------------------------------
__global__ void wfinal_kernel(const float* __restrict__ Wp,
                              const float* __restrict__ out_w,
                              const float* __restrict__ out_b,
                              const float* __restrict__ bp,
                              float* __restrict__ Wf, float* __restrict__ bf) {
  int id = blockIdx.x * 256 + threadIdx.x;  // 64*128 = 8192
  if (id < OUT_C * E) {
    int o = id >> 7, e = id & 127;
    float s = 0.f;
    for (int j = 0; j < E; ++j) s += Wp[o * E + j] * out_w[j * E + e];
    Wf[id] = s;
  }
  if (id < OUT_C) {
    float t = bp[id];
    for (int j = 0; j < E; ++j) t += Wp[id * E + j] * out_b[j];
    bf[id] = t;
  }
}

// ---------------------------------------------------------------------------
// Generic fp32 WMMA GEMM: C[M,N] = A[M,K] * W[N,K]^T + bias[N] (+ PE[row%P,N])
// Block = 128 threads (4 waves); block tile 128(M) x 64(N); each wave owns a
// 32x64 strip -> 8 independent v_wmma_f32_16x16x4_f32 per K=4 step against
// 6 b64 loads, software-pipelined one step ahead.
// A/B frag layout per ISA: lane l<16 carries row base+l with K pair (k0,k0+1);
// lanes 16..31 carry K pair (k0+2,k0+3) -> one b64 load per fragment.
// store_tri: scatter to (B, OUT_C, 3072) triplane layout.
// ---------------------------------------------------------------------------
__global__ __launch_bounds__(128) void wmma_gemm(
    const float* __restrict__ A, const float* __restrict__ W,
    const float* __restrict__ bias, const float* __restrict__ PE, int pe_period,
    float* __restrict__ C, int M, int N, int K, int store_tri) {
  const int lane = threadIdx.x & 31;
  const int wave = threadIdx.x >> 5;
  const int l16 = lane & 15;
  const int half = lane >> 4;
  const int tm = blockIdx.x * 128 + wave * 32;
  const int tn = blockIdx.y * 64;

  int ar0 = tm + l16;       if (ar0 >= M) ar0 = M - 1;  // clamp; stores guarded
  int ar1 = tm + 16 + l16;  if (ar1 >= M) ar1 = M - 1;
  const float* Ap0 = A + (size_t)ar0 * K + 2 * half;
  const float* Ap1 = A + (size_t)ar1 * K + 2 * half;
  const float* Wq0 = W + (size_t)(tn + 0 * 16 + l16) * K + 2 * half;
  const float* Wq1 = W + (size_t)(tn + 1 * 16 + l16) * K + 2 * half;
  const float* Wq2 = W + (size_t)(tn + 2 * 16 + l16) * K + 2 * half;
  const float* Wq3 = W + (size_t)(tn + 3 * 16 + l16) * K + 2 * half;

  v8f acc[8];
#pragma unroll
  for (int i = 0; i < 8; ++i) acc[i] = v8zero();

  // Pipeline prologue: fragments for k0 = 0.
  v2f a0 = *(const v2f*)(Ap0);
  v2f a1 = *(const v2f*)(Ap1);
  v2f b0 = *(const v2f*)(Wq0);
  v2f b1 = *(const v2f*)(Wq1);
  v2f b2 = *(const v2f*)(Wq2);
  v2f b3 = *(const v2f*)(Wq3);

  for (int k0 = 0; k0 + 4 < K; k0 += 4) {
    // Prefetch next K-step while current WMMAs execute.
    v2f na0 = *(const v2f*)(Ap0 + k0 + 4);
    v2f na1 = *(const v2f*)(Ap1 + k0 + 4);
    v2f nb0 = *(const v2f*)(Wq0 + k0 + 4);
    v2f nb1 = *(const v2f*)(Wq1 + k0 + 4);
    v2f nb2 = *(const v2f*)(Wq2 + k0 + 4);
    v2f nb3 = *(const v2f*)(Wq3 + k0 + 4);

    acc[0] = wmma4(a0, b0, acc[0]);
    acc[1] = wmma4(a0, b1, acc[1]);
    acc[2] = wmma4(a0, b2, acc[2]);
    acc[3] = wmma4(a0, b3, acc[3]);
    acc[4] = wmma4(a1, b0, acc[4]);
    acc[5] = wmma4(a1, b1, acc[5]);
    acc[6] = wmma4(a1, b2, acc[6]);
    acc[7] = wmma4(a1, b3, acc[7]);

    a0 = na0; a1 = na1;
    b0 = nb0; b1 = nb1; b2 = nb2; b3 = nb3;
  }
  // Epilogue K-step.
  acc[0] = wmma4(a0, b0, acc[0]);
  acc[1] = wmma4(a0, b1, acc[1]);
  acc[2] = wmma4(a0, b2, acc[2]);
  acc[3] = wmma4(a0, b3, acc[3]);
  acc[4] = wmma4(a1, b0, acc[4]);
  acc[5] = wmma4(a1, b1, acc[5]);
  acc[6] = wmma4(a1, b2, acc[6]);
  acc[7] = wmma4(a1, b3, acc[7]);

#pragma unroll
  for (int r = 0; r < 2; ++r) {
#pragma unroll
    for (int nt = 0; nt < 4; ++nt) {
      int col = tn + nt * 16 + l16;
      float bv = bias[col];
#pragma unroll
      for (int i = 0; i < 8; ++i) {
        int row = tm + r * 16 + i + half * 8;
        if (row < M) {
          float v = acc[r * 4 + nt][i] + bv;
          if (PE) v += PE[(size_t)(row % pe_period) * N + col];
          if (store_tri) {
            int b = row / NQ, n = row - b * NQ;
            C[((size_t)b * OUT_C + col) * NQ + n] = v;
          } else {
            C[(size_t)row * N + col] = v;
          }
        }
      }
    }
  }
}

// ---------------------------------------------------------------------------
// Masked multi-head attention. One wave per (16-query tile, batch, head).
// scores (16x272, keys padded) built with WMMA (K-frags prefetched one tile
// ahead), staged in LDS, softmaxed in-wave (shfl across half-wave column
// partitions), then P @ V as two independent pipelined WMMA chains.
// ---------------------------------------------------------------------------
__global__ __launch_bounds__(32) void attn_kernel(
    const float* __restrict__ q2, const float* __restrict__ k2,
    const float* __restrict__ v2, const unsigned* __restrict__ maskb,
    float* __restrict__ a_out) {
  __shared__ float s[16][274];  // even stride: keeps b64 LDS reads aligned
  const int lane = threadIdx.x & 31;
  const int l16 = lane & 15;
  const int half = lane >> 4;
  const int mt = blockIdx.x;       // 0..191 query tile
  const int bh = blockIdx.y;       // 0..63
  const int bb = bh >> 3;
  const int h = bh & 7;

  // Preload the 16x16 Q tile as four A-fragments (K pairs per half-wave).
  const int qrow = bb * NQ + mt * 16 + l16;
  const float* qp = q2 + (size_t)qrow * E + h * DH + 2 * half;
  v2f qa[4];
#pragma unroll
  for (int f = 0; f < 4; ++f) qa[f] = *(const v2f*)(qp + 4 * f);

  // ---- scores = Q K^T * 0.25, masked to -inf, into LDS -------------------
  const float* kbase = k2 + (size_t)bb * NK * E + h * DH + 2 * half;
  v2f kb[4];
#pragma unroll
  for (int f = 0; f < 4; ++f)  // tile t=0: tok = l16 (< 257)
    kb[f] = *(const v2f*)(kbase + (size_t)l16 * E + 4 * f);

  for (int t = 0; t < 17; ++t) {
    v2f kn[4];
    if (t < 16) {  // prefetch next key tile (uniform branch, EXEC intact)
      int tok = (t + 1) * 16 + l16;
      int tokc = tok > 256 ? 256 : tok;
      const float* kp = kbase + (size_t)tokc * E;
#pragma unroll
      for (int f = 0; f < 4; ++f) kn[f] = *(const v2f*)(kp + 4 * f);
    }
    v8f acc = v8zero();
#pragma unroll
    for (int f = 0; f < 4; ++f) acc = wmma4(qa[f], kb[f], acc);

    int tok = t * 16 + l16;
    const bool inrange = (tok < NK);
    const bool always = (tok == 0);
#pragma unroll
    for (int i = 0; i < 8; ++i) {
      int rloc = i + half * 8;
      bool vis = false;
      if (always) {
        vis = true;
      } else if (inrange) {
        unsigned w =
            maskb[((size_t)(bb * NQ + mt * 16 + rloc)) * 8 + ((tok - 1) >> 5)];
        vis = (w >> ((tok - 1) & 31)) & 1u;
      }
      s[rloc][t * 16 + l16] = vis ? acc[i] * 0.25f : -__builtin_inff();
    }
#pragma unroll
    for (int f = 0; f < 4; ++f) kb[f] = kn[f];
  }
  __syncthreads();

  // ---- softmax over 257 keys; lane pair (l16, l16+16) shares a row -------
  {
    int row = l16;
    float mx = -__builtin_inff();
    for (int c = half; c < NK; c += 2) mx = fmaxf(mx, s[row][c]);
    mx = fmaxf(mx, __shfl_xor(mx, 16, 32));
    float sum = 0.f;
    for (int c = half; c < NK; c += 2) sum += expf(s[row][c] - mx);
    sum += __shfl_xor(sum, 16, 32);
    float inv = 1.0f / sum;
    for (int c = half; c < 272; c += 2)
      s[row][c] = expf(s[row][c] - mx) * inv;  // padded cols were -inf -> 0
  }
  __syncthreads();

  // ---- O = P(16x272) @ V(272x16): two pipelined accumulator chains -------
  const float* vb0 = v2 + (size_t)bb * NK * E + h * DH + l16;
  auto vfrag = [&](int k0) {
    int t0 = k0 + 2 * half;
    int t1 = t0 + 1;
    if (t0 > 256) t0 = 256;  // P is 0 there, value irrelevant
    if (t1 > 256) t1 = 256;
    v2f r;
    r.x = vb0[(size_t)t0 * E];
    r.y = vb0[(size_t)t1 * E];
    return r;
  };
  v8f o0 = v8zero(), o1 = v8zero();
  v2f pa0 = *(const v2f*)&s[l16][0 + 2 * half];
  v2f pa1 = *(const v2f*)&s[l16][4 + 2 * half];
  v2f va0 = vfrag(0);
  v2f va1 = vfrag(4);
  for (int k0 = 0; k0 < 272; k0 += 8) {
    v2f npa0, npa1, nva0, nva1;
    if (k0 + 8 < 272) {  // prefetch next pair of K-steps
      npa0 = *(const v2f*)&s[l16][k0 + 8 + 2 * half];
      npa1 = *(const v2f*)&s[l16][k0 + 12 + 2 * half];
      nva0 = vfrag(k0 + 8);
      nva1 = vfrag(k0 + 12);
    }
    o0 = wmma4(pa0, va0, o0);
    o1 = wmma4(pa1, va1, o1);
    pa0 = npa0; pa1 = npa1; va0 = nva0; va1 = nva1;
  }
  v8f oacc = o0 + o1;

  float* op = a_out + ((size_t)(bb * NQ + mt * 16)) * E + h * DH + l16;
#pragma unroll
  for (int i = 0; i < 8; ++i) op[(size_t)(i + half * 8) * E] = oacc[i];
}

// ---------------------------------------------------------------------------
// Host-side orchestration.
// Workspace layout (floats), total ~36.1 MB:
//   [0 .. 1572864)              tf        (later reused: k_buf, v_buf)
//   [1572864 .. 4718592)        q_buf     (later reused: a_buf)
//   [4718592 .. 5111808)        tri_pe
//   [5111808 .. 5144704)        img_pe
//   [5144704 .. 8290432)        q2
//   [8290432 .. 8553600)        k2
//   [8553600 .. 8816768)        v2
//   [8816768 .. 9013376)        mask bits (unsigned)
//   [9013376 .. 9021568)        W_final
//   [9021568 .. 9021632)        b_final
// ---------------------------------------------------------------------------
extern "C" void kernel_launch(void* const* d_in, const int* in_sizes, int n_in,
                              void* d_out, int out_size, void* d_ws,
                              size_t ws_size, hipStream_t stream) {
  (void)in_sizes; (void)n_in; (void)out_size; (void)ws_size;

  const float* tri   = (const float*)d_in[0];
  const float* img   = (const float*)d_in[1];
  const float* proj  = (const float*)d_in[2];
  const float* Wq    = (const float*)d_in[3];
  const float* bq    = (const float*)d_in[4];
  const float* Wk    = (const float*)d_in[5];
  const float* bk    = (const float*)d_in[6];
  const float* Wv    = (const float*)d_in[7];
  const float* bv    = (const float*)d_in[8];
  const float* in_w  = (const float*)d_in[9];
  const float* in_b  = (const float*)d_in[10];
  const float* out_w = (const float*)d_in[11];
  const float* out_b = (const float*)d_in[12];
  const float* Wp    = (const float*)d_in[13];
  const float* bp    = (const float*)d_in[14];
  float* out = (float*)d_out;

  float* ws = (float*)d_ws;
  const int MQ = BATCH * NQ;   // 24576
  const int MK = BATCH * NK;   // 2056
  float* tf     = ws;
  float* q_buf  = ws + 1572864;
  float* k_buf  = ws;              // reuse tf region after Q path done
  float* v_buf  = ws + 263168;     // still inside old tf region
  float* a_buf  = ws + 1572864;    // reuse q_buf region after Q2
  float* tri_pe = ws + 4718592;
  float* img_pe = ws + 5111808;
  float* q2     = ws + 5144704;
  float* k2     = ws + 8290432;
  float* v2     = ws + 8553600;
  unsigned* maskb = (unsigned*)(ws + 8816768);
  float* Wf     = ws + 9013376;
  float* bf     = ws + 9021568;

  // Independent prep work.
  pe_kernel<<<(NQ * E + 255) / 256, 256, 0, stream>>>(tri_pe, NQ);
  pe_kernel<<<(NK * E + 255) / 256, 256, 0, stream>>>(img_pe, NK);
  mask_kernel<<<dim3(NQ / 256, BATCH), 256, 0, stream>>>(proj, maskb);
  transpose_tf<<<(BATCH * TRI_C * NQ) / 256, 256, 0, stream>>>(tri, tf);
  wfinal_kernel<<<(OUT_C * E + 255) / 256, 256, 0, stream>>>(Wp, out_w, out_b,
                                                             bp, Wf, bf);

  // Q path: q = tf*Wq^T + bq + TRI_PE ; q2 = q*in_w_q^T + in_b_q
  wmma_gemm<<<dim3(MQ / 128, E / 64), 128, 0, stream>>>(
      tf, Wq, bq, tri_pe, NQ, q_buf, MQ, E, TRI_C, 0);
  wmma_gemm<<<dim3(MQ / 128, E / 64), 128, 0, stream>>>(
      q_buf, in_w, in_b, nullptr, 1, q2, MQ, E, E, 0);

  // K path (tf/q_buf now dead; reuse their space).
  wmma_gemm<<<dim3((MK + 127) / 128, E / 64), 128, 0, stream>>>(
      img, Wk, bk, img_pe, NK, k_buf, MK, E, IMG_C, 0);
  wmma_gemm<<<dim3((MK + 127) / 128, E / 64), 128, 0, stream>>>(
      k_buf, in_w + E * E, in_b + E, nullptr, 1, k2, MK, E, E, 0);

  // V path.
  wmma_gemm<<<dim3((MK + 127) / 128, E / 64), 128, 0, stream>>>(
      img, Wv, bv, img_pe, NK, v_buf, MK, E, IMG_C, 0);
  wmma_gemm<<<dim3((MK + 127) / 128, E / 64), 128, 0, stream>>>(
      v_buf, in_w + 2 * E * E, in_b + 2 * E, nullptr, 1, v2, MK, E, E, 0);

  // Attention -> a_buf (B*NQ, E).
  attn_kernel<<<dim3(NQ / 16, BATCH * HEADS), 32, 0, stream>>>(q2, k2, v2,
                                                               maskb, a_buf);

  // Fused out-proj + final proj, stored directly in (B, OUT_C, 3072) layout.
  wmma_gemm<<<dim3(MQ / 128, OUT_C / 64), 128, 0, stream>>>(
      a_buf, Wf, bf, nullptr, 1, out, MQ, OUT_C, E, 1);
}